// DGNNLayer_22660247454026
// MI455X (gfx1250) — compile-verified
//
#include <hip/hip_runtime.h>

// Problem sizes (fixed by the reference)
#define N_NODES 100000
#define N_EDGES 1600000
#define DD      128      // D
#define HH      256      // H == 2*D
#define EPS     1e-5f

typedef __attribute__((ext_vector_type(2))) float v2f;
typedef __attribute__((ext_vector_type(8))) float v8f;

// f32 WMMA: D(16x16 f32) = A(16x4 f32) * B(4x16 f32) + C
__device__ __forceinline__ v8f wmma4(v2f a, v2f b, v8f c) {
  return __builtin_amdgcn_wmma_f32_16x16x4_f32(
      /*neg_a=*/false, a, /*neg_b=*/false, b,
      /*c_mod=*/(short)0, c, /*reuse_a=*/false, /*reuse_b=*/false);
}

// ---------------- workspace layout (float offsets) ----------------
// [agg 100000*128][sum1 256][sq1 256][sum2 256][sq2 256]   <- zeroed each call
// [sc1 256][sh1 256][sc2 256][sh2 256][W1eff 65536][b1eff 256][Y 100000*256]
#define AGG_OFF   ((size_t)0)
#define SUM1_OFF  ((size_t)N_NODES * DD)            // 12,800,000
#define SQ1_OFF   (SUM1_OFF + 256)
#define SUM2_OFF  (SUM1_OFF + 512)
#define SQ2_OFF   (SUM1_OFF + 768)
#define ZERO_CNT  (SUM1_OFF + 1024)                 // floats to zero (div by 4)
#define SC1_OFF   (SUM1_OFF + 1024)
#define SH1_OFF   (SC1_OFF + 256)
#define SC2_OFF   (SC1_OFF + 512)
#define SH2_OFF   (SC1_OFF + 768)
#define W1E_OFF   (SC1_OFF + 1024)
#define B1E_OFF   (W1E_OFF + (size_t)HH * HH)
#define Y_OFF     (B1E_OFF + 256)
// total = Y_OFF + 100000*256 floats  (~154 MB of workspace)

// ---------------- zero kernel ----------------
__global__ void k_zero(float4* __restrict__ p, int n4) {
  int i = blockIdx.x * blockDim.x + threadIdx.x;
  int stride = gridDim.x * blockDim.x;
  float4 z = make_float4(0.f, 0.f, 0.f, 0.f);
  for (; i < n4; i += stride) p[i] = z;
}

// ---------------- SpMM: agg[r] += val * input[c]  (32 lanes/edge, float4/lane) --------
__global__ void k_spmm(const float* __restrict__ x, const int* __restrict__ rows,
                       const int* __restrict__ cols, const float* __restrict__ vals,
                       float* __restrict__ agg) {
  unsigned t = blockIdx.x * blockDim.x + threadIdx.x;
  unsigned e = t >> 5;
  if (e >= N_EDGES) return;
  int lane = t & 31;
  int r = rows[e], c = cols[e];
  float v = vals[e];
  float4 xv = ((const float4*)(x + (size_t)c * DD))[lane];
  float* dst = agg + (size_t)r * DD + lane * 4;
  atomicAdd(dst + 0, v * xv.x);
  atomicAdd(dst + 1, v * xv.y);
  atomicAdd(dst + 2, v * xv.z);
  atomicAdd(dst + 3, v * xv.w);
}

// -------- per-column sum / sumsq of X = [input | agg]  (256 cols) --------
__global__ void k_stats_cat(const float* __restrict__ inp, const float* __restrict__ agg,
                            float* __restrict__ sum, float* __restrict__ sq) {
  int c = threadIdx.x;                       // 0..255
  int rows_per = (N_NODES + gridDim.x - 1) / gridDim.x;
  int r0 = blockIdx.x * rows_per;
  int r1 = min(r0 + rows_per, N_NODES);
  const float* base = (c < DD) ? (inp + c) : (agg + (c - DD));
  float s = 0.f, q = 0.f;
  for (int r = r0; r < r1; ++r) {
    float x = base[(size_t)r * DD];
    s += x; q += x * x;
  }
  atomicAdd(&sum[c], s);
  atomicAdd(&sq[c], q);
}

// -------- per-column sum / sumsq of Y [N,256] --------
__global__ void k_stats_y(const float* __restrict__ Y,
                          float* __restrict__ sum, float* __restrict__ sq) {
  int c = threadIdx.x;
  int rows_per = (N_NODES + gridDim.x - 1) / gridDim.x;
  int r0 = blockIdx.x * rows_per;
  int r1 = min(r0 + rows_per, N_NODES);
  float s = 0.f, q = 0.f;
  for (int r = r0; r < r1; ++r) {
    float x = Y[(size_t)r * HH + c];
    s += x; q += x * x;
  }
  atomicAdd(&sum[c], s);
  atomicAdd(&sq[c], q);
}

// -------- BN finalize: scale = gamma*rsqrt(var+eps), shift = beta - mu*scale ----
__global__ void k_bn_fin(const float* __restrict__ sum, const float* __restrict__ sq,
                         const float* __restrict__ gamma, const float* __restrict__ beta,
                         float* __restrict__ scale, float* __restrict__ shift) {
  int c = threadIdx.x;
  const float invN = 1.0f / (float)N_NODES;
  float mu = sum[c] * invN;
  float var = sq[c] * invN - mu * mu;
  float sc = gamma[c] * rsqrtf(var + EPS);
  scale[c] = sc;
  shift[c] = beta[c] - mu * sc;
}

// -------- fold BN1 into W1: W1eff[k][h] = scale1[k]*W1[k][h] --------
__global__ void k_fold_w1(const float* __restrict__ W1, const float* __restrict__ sc1,
                          float* __restrict__ W1e) {
  int k = blockIdx.x, h = threadIdx.x;
  W1e[(size_t)k * HH + h] = sc1[k] * W1[(size_t)k * HH + h];
}

// -------- b1eff[h] = b1[h] + sum_k shift1[k]*W1[k][h] --------
__global__ void k_fold_b1(const float* __restrict__ W1, const float* __restrict__ sh1,
                          const float* __restrict__ b1, float* __restrict__ b1e) {
  int h = threadIdx.x;
  float s = b1[h];
  for (int k = 0; k < HH; ++k) s += sh1[k] * W1[(size_t)k * HH + h];
  b1e[h] = s;
}

// ---------------- GEMM1: Y[N,256] = X[N,256] @ W1eff + b1eff ----------------
// X cols [0,128) come from `input`, cols [128,256) from `agg`.
// One wave -> 16 rows x 64 cols (4 WMMA accumulators). 25000 waves total.
__global__ void k_gemm1(const float* __restrict__ inp, const float* __restrict__ agg,
                        const float* __restrict__ W, const float* __restrict__ bias,
                        float* __restrict__ Y) {
  const int waveId = blockIdx.x * (blockDim.x >> 5) + (threadIdx.x >> 5);
  const int lane   = threadIdx.x & 31;
  const int rowTile = waveId >> 2;     // 0..6249
  const int colGrp  = waveId & 3;      // 64-col group
  const int m    = lane & 15;
  const int half = lane >> 4;          // lanes 16-31 hold K+2,K+3
  const int n0   = colGrp * 64;
  const int bcol = n0 + m;
  const size_t row = (size_t)(rowTile * 16 + m);

  v8f acc0 = {}, acc1 = {}, acc2 = {}, acc3 = {};

  const float* a0 = inp + row * DD + half * 2;   // k in [0,128)
  const float* a1 = agg + row * DD + half * 2;   // k in [128,256)

#pragma unroll 4
  for (int k = 0; k < DD; k += 4) {
    v2f a = *(const v2f*)(a0 + k);
    const float* w0 = W + (size_t)(k + half * 2) * HH + bcol;
    const float* w1 = w0 + HH;
    v2f wb0 = { w0[0],  w1[0]  };
    v2f wb1 = { w0[16], w1[16] };
    v2f wb2 = { w0[32], w1[32] };
    v2f wb3 = { w0[48], w1[48] };
    acc0 = wmma4(a, wb0, acc0);
    acc1 = wmma4(a, wb1, acc1);
    acc2 = wmma4(a, wb2, acc2);
    acc3 = wmma4(a, wb3, acc3);
  }
#pragma unroll 4
  for (int k = 0; k < DD; k += 4) {
    v2f a = *(const v2f*)(a1 + k);
    const float* w0 = W + (size_t)(DD + k + half * 2) * HH + bcol;
    const float* w1 = w0 + HH;
    v2f wb0 = { w0[0],  w1[0]  };
    v2f wb1 = { w0[16], w1[16] };
    v2f wb2 = { w0[32], w1[32] };
    v2f wb3 = { w0[48], w1[48] };
    acc0 = wmma4(a, wb0, acc0);
    acc1 = wmma4(a, wb1, acc1);
    acc2 = wmma4(a, wb2, acc2);
    acc3 = wmma4(a, wb3, acc3);
  }

  const int c = n0 + m;
  float bv0 = bias[c], bv1 = bias[c + 16], bv2 = bias[c + 32], bv3 = bias[c + 48];
#pragma unroll
  for (int j = 0; j < 8; ++j) {
    size_t r = (size_t)(rowTile * 16 + j + half * 8);
    float* yr = Y + r * HH + c;
    yr[0]  = acc0[j] + bv0;
    yr[16] = acc1[j] + bv1;
    yr[32] = acc2[j] + bv2;
    yr[48] = acc3[j] + bv3;
  }
}

// ---------------- GEMM2: out[N,128] = relu(Y*sc2+sh2) @ W2 + b2 ----------------
// One wave -> 16 rows x 64 cols. 12500 waves total.
__global__ void k_gemm2(const float* __restrict__ Y, const float* __restrict__ sc,
                        const float* __restrict__ sh, const float* __restrict__ W2,
                        const float* __restrict__ bias, float* __restrict__ out) {
  const int waveId = blockIdx.x * (blockDim.x >> 5) + (threadIdx.x >> 5);
  if (waveId >= (N_NODES / 16) * 2) return;
  const int lane   = threadIdx.x & 31;
  const int rowTile = waveId >> 1;
  const int colGrp  = waveId & 1;
  const int m    = lane & 15;
  const int half = lane >> 4;
  const int n0   = colGrp * 64;
  const int bcol = n0 + m;
  const size_t row = (size_t)(rowTile * 16 + m);

  v8f acc0 = {}, acc1 = {}, acc2 = {}, acc3 = {};
  const float* yrow = Y + row * HH + half * 2;

#pragma unroll 4
  for (int k = 0; k < HH; k += 4) {
    int kk = k + half * 2;
    v2f yv = *(const v2f*)(yrow + k);
    v2f a;
    a.x = fmaxf(fmaf(yv.x, sc[kk],     sh[kk]),     0.0f);
    a.y = fmaxf(fmaf(yv.y, sc[kk + 1], sh[kk + 1]), 0.0f);
    const float* w0 = W2 + (size_t)kk * DD + bcol;
    const float* w1 = w0 + DD;
    v2f wb0 = { w0[0],  w1[0]  };
    v2f wb1 = { w0[16], w1[16] };
    v2f wb2 = { w0[32], w1[32] };
    v2f wb3 = { w0[48], w1[48] };
    acc0 = wmma4(a, wb0, acc0);
    acc1 = wmma4(a, wb1, acc1);
    acc2 = wmma4(a, wb2, acc2);
    acc3 = wmma4(a, wb3, acc3);
  }

  const int c = n0 + m;
  float bv0 = bias[c], bv1 = bias[c + 16], bv2 = bias[c + 32], bv3 = bias[c + 48];
#pragma unroll
  for (int j = 0; j < 8; ++j) {
    size_t r = (size_t)(rowTile * 16 + j + half * 8);
    float* o = out + r * DD + c;
    o[0]  = acc0[j] + bv0;
    o[16] = acc1[j] + bv1;
    o[32] = acc2[j] + bv2;
    o[48] = acc3[j] + bv3;
  }
}

// ---------------- launcher ----------------
extern "C" void kernel_launch(void* const* d_in, const int* in_sizes, int n_in,
                              void* d_out, int out_size, void* d_ws, size_t ws_size,
                              hipStream_t stream) {
  const float* input    = (const float*)d_in[0];
  const int*   adj_rows = (const int*)  d_in[1];
  const int*   adj_cols = (const int*)  d_in[2];
  const float* adj_vals = (const float*)d_in[3];
  const float* gamma1   = (const float*)d_in[4];
  const float* beta1    = (const float*)d_in[5];
  const float* W1       = (const float*)d_in[6];
  const float* b1       = (const float*)d_in[7];
  const float* gamma2   = (const float*)d_in[8];
  const float* beta2    = (const float*)d_in[9];
  const float* W2       = (const float*)d_in[10];
  const float* b2       = (const float*)d_in[11];
  float* out = (float*)d_out;
  float* ws  = (float*)d_ws;

  float* agg  = ws + AGG_OFF;
  float* sum1 = ws + SUM1_OFF;
  float* sq1  = ws + SQ1_OFF;
  float* sum2 = ws + SUM2_OFF;
  float* sq2  = ws + SQ2_OFF;
  float* sc1  = ws + SC1_OFF;
  float* sh1  = ws + SH1_OFF;
  float* sc2  = ws + SC2_OFF;
  float* sh2  = ws + SH2_OFF;
  float* W1e  = ws + W1E_OFF;
  float* b1e  = ws + B1E_OFF;
  float* Y    = ws + Y_OFF;

  // 1) zero agg + stat accumulators
  k_zero<<<2048, 256, 0, stream>>>((float4*)ws, (int)(ZERO_CNT / 4));
  // 2) SpMM (edge-parallel, atomic scatter-add; everything L2-resident)
  k_spmm<<<(N_EDGES * 32) / 256, 256, 0, stream>>>(input, adj_rows, adj_cols, adj_vals, agg);
  // 3) BN1 stats over [input | agg]
  k_stats_cat<<<400, 256, 0, stream>>>(input, agg, sum1, sq1);
  // 4) BN1 finalize
  k_bn_fin<<<1, 256, 0, stream>>>(sum1, sq1, gamma1, beta1, sc1, sh1);
  // 5) fold BN1 into W1 / b1
  k_fold_w1<<<HH, HH, 0, stream>>>(W1, sc1, W1e);
  k_fold_b1<<<1, HH, 0, stream>>>(W1, sh1, b1, b1e);
  // 6) GEMM1 (WMMA f32): 6250*4 = 25000 waves, 8 waves/block
  k_gemm1<<<3125, 256, 0, stream>>>(input, agg, W1e, b1e, Y);
  // 7) BN2 stats over Y
  k_stats_y<<<400, 256, 0, stream>>>(Y, sum2, sq2);
  // 8) BN2 finalize
  k_bn_fin<<<1, 256, 0, stream>>>(sum2, sq2, gamma2, beta2, sc2, sh2);
  // 9) GEMM2 (WMMA f32, fused BN2+ReLU on A): 12500 waves
  k_gemm2<<<1563, 256, 0, stream>>>(Y, sc2, sh2, W2, b2, out);
}